// LocallyConnected2dV1_32323923870271
// MI455X (gfx1250) — compile-verified
//
#include <hip/hip_runtime.h>

// LocallyConnected2d: out[h,w] = sum_{kh,kw} x[h+kh, w+kw] * W[h, w, kh*16+kw]
// Memory-bound (W = 253 MB streamed once; floor ~11us @ 23.3 TB/s).
// Strategy: one wave32 computes 16 outputs (same h, consecutive w) via the
// diagonal of 64 chained V_WMMA_F32_16X16X4_F32 accumulations. W is streamed
// with non-temporal b64 loads (sequential per lane); x is cached/resident.

typedef float v2f __attribute__((ext_vector_type(2)));
typedef float v8f __attribute__((ext_vector_type(8)));

#define IN_H 512
#define IN_W 512
#define KH 16
#define KW 16
#define OUT_H 497
#define OUT_W 497
#define PATCH (KH * KW)          // 256
#define WTILES 32                // ceil(497/16)
#define TOTAL_TILES (OUT_H * WTILES)
#define WAVES_PER_BLOCK 8

__global__ __launch_bounds__(256) void lc2d_wmma_kernel(
    const float* __restrict__ x,   // (512, 512)
    const float* __restrict__ W,   // (497, 497, 256)
    float* __restrict__ out)       // (497, 497)
{
    const int lane = threadIdx.x & 31;
    const int wave = threadIdx.x >> 5;
    const int tile = blockIdx.x * WAVES_PER_BLOCK + wave;   // wave-uniform
    if (tile >= TOTAL_TILES) return;                        // whole-wave exit

    const int h  = tile / WTILES;          // output row (uniform)
    const int w0 = (tile % WTILES) * 16;   // first output col of tile (uniform)

    // Per-lane matrix coordinates.
    // A (16x4 f32): lanes 0-15 hold A[m][0..1], lanes 16-31 hold A[m][2..3].
    // B (4x16 f32): lanes 0-15 hold B[0..1][n], lanes 16-31 hold B[2..3][n].
    const int m  = lane & 15;                   // output index within tile
    const int hi = (lane >= 16) ? 2 : 0;        // K sub-offset for this half-wave
    int wn = w0 + m;                            // clamp OOB columns (store is masked)
    if (wn > OUT_W - 1) wn = OUT_W - 1;

    // Per-lane streaming base into this output's private 256-float weight vector.
    const float* wp = W + ((size_t)h * OUT_W + wn) * PATCH + hi;   // 8B aligned
    // Per-lane base into x for B: col = wn + hi + kwb (+0/+1), row = h + kh.
    const float* xp = x + (size_t)h * IN_W + wn + hi;

    v8f c0 = {};
    v8f c1 = {};

    #pragma unroll
    for (int kh = 0; kh < KH; ++kh) {
        #pragma unroll
        for (int q = 0; q < 4; ++q) {          // kwb = 4*q, K-chunk pb = kh*16 + 4*q
            // A chunk: W[h][wn][pb + hi .. pb + hi + 1]  (non-temporal stream)
            v2f a = __builtin_nontemporal_load(
                (const v2f*)(wp + kh * KW + q * 4));
            // B chunk: x[h+kh][wn + hi + 4q (+1)] (hot in cache, 4B aligned)
            v2f b;
            b.x = xp[kh * IN_W + q * 4];
            b.y = xp[kh * IN_W + q * 4 + 1];
            if (kh & 1)
                c1 = __builtin_amdgcn_wmma_f32_16x16x4_f32(
                        false, a, false, b, (short)0, c1, false, false);
            else
                c0 = __builtin_amdgcn_wmma_f32_16x16x4_f32(
                        false, a, false, b, (short)0, c0, false, false);
        }
    }

    v8f c = c0 + c1;

    // Diagonal extraction: D[n][n] sits at (VGPR n, lane n) for n<8 and at
    // (VGPR n-8, lane n+16) for n>=8.  Active lanes: 0-7 and 24-31.
    const int s = lane & 7;
    float d = c[0];
    if (s == 1) d = c[1];
    if (s == 2) d = c[2];
    if (s == 3) d = c[3];
    if (s == 4) d = c[4];
    if (s == 5) d = c[5];
    if (s == 6) d = c[6];
    if (s == 7) d = c[7];

    const bool active = (lane < 8) || (lane >= 24);
    const int n = (lane < 16) ? lane : (lane - 16);
    const int w = w0 + n;
    if (active && w < OUT_W) {
        out[(size_t)h * OUT_W + w] = d;
    }
}

extern "C" void kernel_launch(void* const* d_in, const int* in_sizes, int n_in,
                              void* d_out, int out_size, void* d_ws, size_t ws_size,
                              hipStream_t stream) {
    const float* x = (const float*)d_in[0];   // 512*512 f32
    const float* W = (const float*)d_in[1];   // 497*497*256 f32
    float* out = (float*)d_out;               // 497*497 f32
    (void)in_sizes; (void)n_in; (void)out_size; (void)d_ws; (void)ws_size;

    const int blocks = (TOTAL_TILES + WAVES_PER_BLOCK - 1) / WAVES_PER_BLOCK;
    lc2d_wmma_kernel<<<blocks, 256, 0, stream>>>(x, W, out);
}